// SpatialLoss_80857054315266
// MI455X (gfx1250) — compile-verified
//
#include <hip/hip_runtime.h>
#include <math.h>
#include <stdint.h>

// Problem constants (from the reference)
#define BT     512      // B*T frames
#define NQ     900      // queries
#define NG     24       // ground-truth boxes
#define M1     (NQ + 1) // JV column arrays are 1-based
#define NTHR   256      // 8 waves of 32
#define CSTR   904      // cost row stride (padded)

typedef __attribute__((ext_vector_type(2))) float v2f;
typedef __attribute__((ext_vector_type(4))) float v4f;
typedef __attribute__((ext_vector_type(8))) float v8f;
typedef int v4i_ __attribute__((vector_size(16)));

#if __has_builtin(__builtin_amdgcn_wmma_f32_16x16x4_f32)
#define USE_WMMA_RED 1
#endif

#if __has_builtin(__builtin_amdgcn_global_load_async_to_lds_b128)
#define HAVE_ASYNC_LDS 1
#endif

__device__ __forceinline__ void async_copy16(const void* g, void* l) {
#ifdef HAVE_ASYNC_LDS
    __builtin_amdgcn_global_load_async_to_lds_b128(
        (__attribute__((address_space(1))) v4i_*)(uintptr_t)g,
        (__attribute__((address_space(3))) v4i_*)l,
        0, 0);
#else
    *(v4f*)l = *(const v4f*)g;
#endif
}

__device__ __forceinline__ void async_join() {
#ifdef HAVE_ASYNC_LDS
#if __has_builtin(__builtin_amdgcn_s_wait_asynccnt)
    __builtin_amdgcn_s_wait_asynccnt(0);
#else
    asm volatile("s_wait_asynccnt 0" ::: "memory");
#endif
#endif
}

__device__ __forceinline__ float softplusf(float x) {
    return fmaxf(x, 0.0f) + log1pf(expf(-fabsf(x)));
}

__device__ __forceinline__ float sigmoidf(float x) {
    return 1.0f / (1.0f + expf(-x));
}

__device__ __forceinline__ float giou_box(float ax, float ay, float aw, float ah,
                                          float bx, float by, float bw, float bh) {
    float ax1 = ax - aw * 0.5f, ay1 = ay - ah * 0.5f;
    float ax2 = ax + aw * 0.5f, ay2 = ay + ah * 0.5f;
    float bx1 = bx - bw * 0.5f, by1 = by - bh * 0.5f;
    float bx2 = bx + bw * 0.5f, by2 = by + bh * 0.5f;
    float ix1 = fmaxf(ax1, bx1), iy1 = fmaxf(ay1, by1);
    float ix2 = fminf(ax2, bx2), iy2 = fminf(ay2, by2);
    float inter = fmaxf(ix2 - ix1, 0.0f) * fmaxf(iy2 - iy1, 0.0f);
    float area_a = (ax2 - ax1) * (ay2 - ay1);
    float area_b = (bx2 - bx1) * (by2 - by1);
    float uni = area_a + area_b - inter;
    float iou = inter / fmaxf(uni, 1e-6f);
    float ex1 = fminf(ax1, bx1), ey1 = fminf(ay1, by1);
    float ex2 = fmaxf(ax2, bx2), ey2 = fmaxf(ay2, by2);
    float enc = (ex2 - ex1) * (ey2 - ey1);
    return iou - (enc - uni) / fmaxf(enc, 1e-6f);
}

// Block-wide sum over 256 threads; wave totals via V_WMMA_F32_16X16X4_F32.
// Result valid on thread 0 only.
__device__ float block_sum256(float v, float* red) {
    float s;
#ifdef USE_WMMA_RED
    v2f a; a[0] = v;    a[1] = 0.0f;
    v2f b; b[0] = 1.0f; b[1] = 1.0f;
    v8f c = {0.f, 0.f, 0.f, 0.f, 0.f, 0.f, 0.f, 0.f};
    v8f d = __builtin_amdgcn_wmma_f32_16x16x4_f32(
        false, a, false, b, (short)0, c, false, false);
    s = ((d[0] + d[1]) + (d[2] + d[3])) + ((d[4] + d[5]) + (d[6] + d[7]));
    s += __shfl_xor(s, 16, 32);
#else
    s = v;
    for (int off = 16; off > 0; off >>= 1) s += __shfl_down(s, off, 32);
#endif
    int wid = threadIdx.x >> 5, lane = threadIdx.x & 31;
    if (lane == 0) red[wid] = s;
    __syncthreads();
    float tot = 0.0f;
    if (threadIdx.x == 0) {
        for (int w = 0; w < 8; ++w) tot += red[w];
    }
    __syncthreads();
    return tot;
}

__global__ void zero_out_kernel(float* out) {
    if (threadIdx.x == 0 && blockIdx.x == 0) out[0] = 0.0f;
}

__global__ __launch_bounds__(NTHR)
void spatial_loss_kernel(const float* __restrict__ bboxes,
                         const float* __restrict__ scores,
                         const float* __restrict__ gt,
                         float* __restrict__ out) {
    __shared__ float s_cost[NG * CSTR];   // 86,784 B  precomputed cost matrix
    __shared__ float s_bbox[NQ * 4];      // 14,400 B
    __shared__ float s_prob[NQ];
    __shared__ float s_gt[NG * 4];
    __shared__ float s_v[M1];
    __shared__ float s_minv[M1];
    __shared__ int   s_way[M1];
    __shared__ int   s_p[M1];
    __shared__ unsigned char s_used[M1];
    __shared__ float s_u[NG + 1];
    __shared__ int   s_match[NG];
    __shared__ unsigned char s_tgt[NQ];
    __shared__ float s_red[8];
    __shared__ float s_redf[8];
    __shared__ int   s_redi[8];
    __shared__ float s_ml1[NG];
    __shared__ float s_mg[NG];
    __shared__ int   s_j0, s_i0, s_k, s_done;
    __shared__ float s_delta;

    const int tid = threadIdx.x;
    const int f = blockIdx.x;
    const float INF = __builtin_inff();

    const float* bb = bboxes + (size_t)f * NQ * 4;
    const float* sc = scores + (size_t)f * NQ;
    const float* gb = gt + (size_t)f * NG * 4;

    // ---- stage frame data into LDS (async-to-LDS path on gfx1250) ----
    for (int i = tid; i < NQ; i += NTHR)
        async_copy16(bb + i * 4, &s_bbox[i * 4]);
    if (tid < NG)
        async_copy16(gb + tid * 4, &s_gt[tid * 4]);
    for (int q = tid; q < NQ; q += NTHR)      // overlaps with async staging
        s_prob[q] = sigmoidf(sc[q]);
    for (int j = tid; j < M1; j += NTHR) { s_p[j] = 0; s_v[j] = 0.0f; }
    if (tid <= NG) s_u[tid] = 0.0f;
    async_join();
    __syncthreads();

    // ---- precompute full 24x900 cost matrix into LDS ----
    for (int t = tid; t < NG * NQ; t += NTHR) {
        const int g = t / NQ;
        const int q = t - g * NQ;
        const float qx = s_bbox[q * 4 + 0], qy = s_bbox[q * 4 + 1];
        const float qw = s_bbox[q * 4 + 2], qh = s_bbox[q * 4 + 3];
        const float gx = s_gt[g * 4 + 0], gy = s_gt[g * 4 + 1];
        const float gw = s_gt[g * 4 + 2], gh = s_gt[g * 4 + 3];
        const float l1 = fabsf(qx - gx) + fabsf(qy - gy) +
                         fabsf(qw - gw) + fabsf(qh - gh);
        const float gi = giou_box(qx, qy, qw, qh, gx, gy, gw, gh);
        s_cost[g * CSTR + q] = -s_prob[q] + 5.0f * l1 - 2.0f * gi;
    }
    __syncthreads();

    // ---- Jonker-Volgenant: rows = GT (24), cols = queries (900) ----
    for (int i = 1; i <= NG; ++i) {
        for (int j = tid; j < M1; j += NTHR) { s_minv[j] = INF; s_used[j] = 0; }
        if (tid == 0) { s_p[0] = i; s_j0 = 0; s_done = 0; }
        __syncthreads();

        for (;;) {
            if (tid == 0) { s_used[s_j0] = 1; s_i0 = s_p[s_j0]; }
            __syncthreads();

            const int i0 = s_i0;
            const int j0cur = s_j0;
            const float ui0 = s_u[i0];
            const float* crow = &s_cost[(i0 - 1) * CSTR];

            float bestv = INF;
            int besti = 0x7fffffff;
            for (int j = 1 + tid; j <= NQ; j += NTHR) {
                if (!s_used[j]) {
                    const float cur = crow[j - 1] - ui0 - s_v[j];
                    float mv = s_minv[j];
                    if (cur < mv) { mv = cur; s_minv[j] = cur; s_way[j] = j0cur; }
                    if (mv < bestv || (mv == bestv && j < besti)) { bestv = mv; besti = j; }
                }
            }
            // block argmin (tie -> lowest column index, matching np.argmin)
            for (int off = 16; off > 0; off >>= 1) {
                float ov = __shfl_down(bestv, off, 32);
                int   oi = __shfl_down(besti, off, 32);
                if (ov < bestv || (ov == bestv && oi < besti)) { bestv = ov; besti = oi; }
            }
            {
                int wid = tid >> 5, lane = tid & 31;
                if (lane == 0) { s_redf[wid] = bestv; s_redi[wid] = besti; }
            }
            __syncthreads();
            if (tid == 0) {
                float fv = s_redf[0]; int fi = s_redi[0];
                for (int w = 1; w < 8; ++w) {
                    float ov = s_redf[w]; int oi = s_redi[w];
                    if (ov < fv || (ov == fv && oi < fi)) { fv = ov; fi = oi; }
                }
                s_delta = fv; s_k = fi;
            }
            __syncthreads();

            const float delta = s_delta;
            const int k = s_k;
            for (int j = tid; j <= NQ; j += NTHR) {
                if (s_used[j]) {
                    s_u[s_p[j]] += delta;   // distinct rows per used column
                    s_v[j] -= delta;
                } else {
                    s_minv[j] -= delta;
                }
            }
            __syncthreads();
            if (tid == 0) {
                s_j0 = k;
                if (s_p[k] == 0) s_done = 1;
            }
            __syncthreads();
            if (s_done) break;
        }
        if (tid == 0) {          // augment along the alternating path
            int j0 = s_j0;
            while (j0) {
                int j1 = s_way[j0];
                s_p[j0] = s_p[j1];
                j0 = j1;
            }
        }
        __syncthreads();
    }

    // ---- extract match (query index per GT) and target mask ----
    for (int j = 1 + tid; j <= NQ; j += NTHR) {
        int pr = s_p[j];
        if (pr > 0) s_match[pr - 1] = j - 1;
    }
    for (int q = tid; q < NQ; q += NTHR) s_tgt[q] = 0;
    __syncthreads();
    if (tid < NG) s_tgt[s_match[tid]] = 1;
    __syncthreads();

    // ---- focal classification loss over all queries ----
    float ps = 0.0f;
    for (int q = tid; q < NQ; q += NTHR) {
        const float x = sc[q];
        const float t = (float)s_tgt[q];
        const float ce = t * softplusf(-x) + (1.0f - t) * softplusf(x);
        const float pr = s_prob[q];
        const float pt = pr * t + (1.0f - pr) * (1.0f - t);
        const float om = 1.0f - pt;
        const float fw = (0.25f * t + 0.75f * (1.0f - t)) * om * om;
        ps += fw * ce;
    }
    const float cls_tot = block_sum256(ps, s_red);  // valid on thread 0

    // ---- matched L1 + GIoU losses ----
    if (tid < NG) {
        const int q = s_match[tid];
        const float qx = s_bbox[q * 4 + 0], qy = s_bbox[q * 4 + 1];
        const float qw = s_bbox[q * 4 + 2], qh = s_bbox[q * 4 + 3];
        const float gx = s_gt[tid * 4 + 0], gy = s_gt[tid * 4 + 1];
        const float gw = s_gt[tid * 4 + 2], gh = s_gt[tid * 4 + 3];
        s_ml1[tid] = fabsf(qx - gx) + fabsf(qy - gy) + fabsf(qw - gw) + fabsf(qh - gh);
        s_mg[tid]  = giou_box(qx, qy, qw, qh, gx, gy, gw, gh);
    }
    __syncthreads();

    if (tid == 0) {
        float l1t = 0.0f, giout = 0.0f;
        for (int g2 = 0; g2 < NG; ++g2) { l1t += s_ml1[g2]; giout += s_mg[g2]; }
        const float loss_cls = cls_tot / (float)NQ;
        const float loss_l1  = l1t / (float)(NG * 4);
        const float loss_g   = ((float)NG - giout) / (float)NG;
        const float frame = loss_cls + 5.0f * loss_l1 + 2.0f * loss_g;
        atomicAdd(out, frame * (1.0f / (float)BT));
    }
}

extern "C" void kernel_launch(void* const* d_in, const int* in_sizes, int n_in,
                              void* d_out, int out_size, void* d_ws, size_t ws_size,
                              hipStream_t stream) {
    (void)in_sizes; (void)n_in; (void)out_size; (void)d_ws; (void)ws_size;
    const float* bboxes = (const float*)d_in[0];
    const float* scores = (const float*)d_in[1];
    const float* gtb    = (const float*)d_in[2];
    float* out = (float*)d_out;

    zero_out_kernel<<<1, 32, 0, stream>>>(out);
    spatial_loss_kernel<<<BT, NTHR, 0, stream>>>(bboxes, scores, gtb, out);
}